// OriginalMACEModel_29815662969341
// MI455X (gfx1250) — compile-verified
//
#include <hip/hip_runtime.h>
#include <hip/hip_bf16.h>
#include <math.h>

// ---------------------------------------------------------------------------
// MACE-style equivariant GNN forward for MI455X (gfx1250, wave32, WMMA + TDM).
//
// Input layout assumed (setup_inputs() dict order, params pytree in insertion
// order):
//   d_in[0]      pos                [N,3] f32
//   d_in[1..43]  params leaves:
//     0 node_embed(1,64) 1 ae_w(64,1) 2 lin_up1(64,64)
//     3..6 rmlp1: (8,64)(64,64)(64,64)(64,192)
//     7..9 lin1{0,1,2}(64,64)  10..12 skip1  13..15 p1_w1(64,)
//     16..18 p1_w2  19..21 p1_w3  22..24 p1_lin  25 read1(64,1) 26 sc2(64,64)
//     27..29 lin_up2  30..33 rmlp2: (8,64)(64,64)(64,64)(64,704)
//     34..36 lin2  37 p2_w1(64,) 38 p2_w2(3,64) 39 p2_w3(11,64) 40 p2_lin
//     41 read2_w1(64,64) 42 read2_w2(64,1)
//   d_in[44] atoms (unused: all zero, one-hot width 1)
//   d_in[45] edge_index [2,E] i32 (snd = ptr, rcv = ptr+E)
//   d_in[46] batch_ids [N] i32
// Output: d_out = 16 f32 graph energies.
// ---------------------------------------------------------------------------

#define NNODES 16000
#define NEDGES 128000
#define NGRAPH 16

typedef float v2f __attribute__((ext_vector_type(2)));
typedef float v8f __attribute__((ext_vector_type(8)));
typedef unsigned int v4u __attribute__((ext_vector_type(4)));
typedef unsigned int v8u __attribute__((ext_vector_type(8)));

__device__ __forceinline__ float siluf(float v) { return v / (1.0f + expf(-v)); }
__device__ __forceinline__ bool validl(int a, int b, int c) {
  int d = a - b; if (d < 0) d = -d;
  return d <= c && c <= (a + b) && (((a + b + c) & 1) == 0);
}

// ===================== CDNA5 TDM / async-LDS helpers ======================
// TDM load of a 64x64 f32 row-major matrix from global into LDS with a
// 68-float destination row stride (pad_enable: +4 dwords after every 64).
// D# layout per CDNA5 ISA.
__device__ __forceinline__ void tdm_load_w64(const float* W, void* ldsdst) {
  unsigned lds = (unsigned)(uintptr_t)ldsdst;  // low 32 bits = LDS byte addr
  unsigned long long ga = (unsigned long long)(uintptr_t)W;
  v4u g0;
  g0[0] = 1u;                                             // count=1, user D#
  g0[1] = lds;                                            // lds_addr
  g0[2] = (unsigned)ga;                                   // global_addr[31:0]
  g0[3] = ((unsigned)(ga >> 32) & 0x01FFFFFFu) | 0x80000000u;  // addr hi, type=2
  v8u g1;
  // data_size=2 (4B), pad_enable=1, pad_interval=5 (64 dw), pad_amount=3 (4 dw)
  g1[0] = (2u << 16) | (1u << 20) | (5u << 22) | (3u << 25);
  g1[1] = 64u << 16;   // tensor_dim0 = 64 (bits 79:48, low half here)
  g1[2] = 64u << 16;   // tensor_dim1 = 64 (bits 111:80, low half here)
  g1[3] = 64u << 16;   // tile_dim0 = 64 (bits 127:112)
  g1[4] = 64u;         // tile_dim1 = 64 (bits 143:128), tile_dim2 = 0
  g1[5] = 64u;         // tensor_dim0_stride = 64 (bits 207:160, low 32)
  g1[6] = 0u;          // stride0 hi=0, tensor_dim1_stride lo=0
  g1[7] = 0u;
  asm volatile("tensor_load_to_lds %0, %1" :: "s"(g0), "s"(g1) : "memory");
}

// Per-lane async global->LDS dword copy (tracked with ASYNCcnt).
__device__ __forceinline__ void async_gather_b32(unsigned lds_off,
                                                 const float* gptr) {
  asm volatile("global_load_async_to_lds_b32 %0, %1, off"
               :: "v"(lds_off), "v"(gptr) : "memory");
}
__device__ __forceinline__ void async_wait0() {
  asm volatile("s_wait_asynccnt 0x0" ::: "memory");
}

// ============================ CG table generation ==========================
__device__ __forceinline__ double d_fact(int n) {
  const double t[9] = {1., 1., 2., 6., 24., 120., 720., 5040., 40320.};
  return t[n];
}

__device__ double d_su2cg(int j1, int m1, int j2, int m2, int j3, int m3) {
  if (m3 != m1 + m2) return 0.0;
  int vmin = -j1 + j2 + m3;
  if (-j1 + m1 > vmin) vmin = -j1 + m1;
  if (0 > vmin) vmin = 0;
  int vmax = j2 + j3 + m1;
  if (j3 - j1 + j2 < vmax) vmax = j3 - j1 + j2;
  if (j3 + m3 < vmax) vmax = j3 + m3;
  double cst = sqrt((2.0 * j3 + 1.0) *
      (d_fact(j3 + j1 - j2) * d_fact(j3 - j1 + j2) * d_fact(j1 + j2 - j3) *
       d_fact(j3 + m3) * d_fact(j3 - m3)) /
      (d_fact(j1 + j2 + j3 + 1) * d_fact(j1 - m1) * d_fact(j1 + m1) *
       d_fact(j2 - m2) * d_fact(j2 + m2)));
  double S = 0.0;
  for (int v = vmin; v <= vmax; ++v) {
    double sgn = ((v + j2 + m2) & 1) ? -1.0 : 1.0;
    S += sgn / d_fact(v) * d_fact(j2 + j3 + m1 - v) * d_fact(j1 - m1 + v) /
         d_fact(j3 - j1 + j2 - v) / d_fact(j3 + m3 - v) / d_fact(v + j1 - j2 - m3);
  }
  return cst * S;
}

__device__ void d_qmat(int l, double qr[5][5], double qi[5][5]) {
  for (int a = 0; a < 5; a++)
    for (int b = 0; b < 5; b++) { qr[a][b] = 0.0; qi[a][b] = 0.0; }
  const double iv = 0.70710678118654752440;
  for (int m = -l; m < 0; m++) { qr[l + m][l - m] = iv; qi[l + m][l + m] = -iv; }
  qr[l][l] = 1.0;
  for (int m = 1; m <= l; m++) {
    double s = (m & 1) ? -1.0 : 1.0;
    qr[l + m][l + m] = s * iv;
    qi[l + m][l - m] = s * iv;
  }
  // multiply by (-i)^l
  if (l == 1) {
    for (int a = 0; a < 5; a++) for (int b = 0; b < 5; b++) {
      double re = qr[a][b], im = qi[a][b];
      qr[a][b] = im; qi[a][b] = -re;
    }
  } else if (l == 2) {
    for (int a = 0; a < 5; a++) for (int b = 0; b < 5; b++) {
      qr[a][b] = -qr[a][b]; qi[a][b] = -qi[a][b];
    }
  }
}

// One thread per (l1,l2,l3) triple -> real-basis CG tensor into ws[27][125].
__global__ void cg_init_kernel(float* __restrict__ CG) {
  int t = threadIdx.x;
  if (t >= 27) return;
  int l1 = t / 9, l2 = (t / 3) % 3, l3 = t % 3;
  float* out = CG + t * 125;
  for (int i = 0; i < 125; i++) out[i] = 0.f;
  if (!validl(l1, l2, l3)) return;
  double q1r[5][5], q1i[5][5], q2r[5][5], q2i[5][5], q3r[5][5], q3i[5][5];
  d_qmat(l1, q1r, q1i); d_qmat(l2, q2r, q2i); d_qmat(l3, q3r, q3i);
  int d1 = 2 * l1 + 1, d2 = 2 * l2 + 1, d3 = 2 * l3 + 1;
  for (int j = 0; j < d1; j++)
    for (int lc = 0; lc < d2; lc++)
      for (int nc = 0; nc < d3; nc++) {
        double are = 0.0;
        for (int i = 0; i < d1; i++)
          for (int k = 0; k < d2; k++)
            for (int m = 0; m < d3; m++) {
              int m1 = i - l1, m2 = k - l2, m3 = m - l3;
              if (m1 + m2 != m3) continue;
              double cg = d_su2cg(l1, m1, l2, m2, l3, m3);
              if (cg == 0.0) continue;
              double ar = q1r[i][j], ai = q1i[i][j];
              double br = q2r[k][lc], bi = q2i[k][lc];
              double cr = q3r[m][nc], ci = -q3i[m][nc];  // conj(q3)
              double abr = ar * br - ai * bi, abi = ar * bi + ai * br;
              are += (abr * cr - abi * ci) * cg;
            }
        out[(j * 5 + lc) * 5 + nc] = (float)are;
      }
}

// ======================= tiny node-constant precompute =====================
// atoms all-zero + one-hot width 1 => h0 row == node_embed row for all nodes.
__global__ void node_embed_kernel(const float* __restrict__ node_embed,
                                  const float* __restrict__ lin_up1,
                                  const float* __restrict__ ae_w,
                                  float* __restrict__ hvec,
                                  float* __restrict__ dot_ae) {
  int f = threadIdx.x;
  float s = 0.f;
  for (int c = 0; c < 64; c++) s += node_embed[c] * lin_up1[c * 64 + f];
  hvec[f] = s;
  if (f == 0) {
    float d = 0.f;
    for (int c = 0; c < 64; c++) d += node_embed[c] * ae_w[c];
    dot_ae[0] = d;
  }
}

// =========================== edge geometry ================================
__global__ void edge_geom_kernel(const float* __restrict__ pos,
                                 const int* __restrict__ ei,
                                 float* __restrict__ Y, float* __restrict__ eb,
                                 int E) {
  int e = blockIdx.x * blockDim.x + threadIdx.x;
  if (e >= E) return;
  int s = ei[e], r = ei[E + e];
  float dx = pos[s * 3 + 0] - pos[r * 3 + 0];
  float dy = pos[s * 3 + 1] - pos[r * 3 + 1];
  float dz = pos[s * 3 + 2] - pos[r * 3 + 2];
  float rn = sqrtf(dx * dx + dy * dy + dz * dz);
  float inv = 1.f / rn;
  float x = dx * inv, y = dy * inv, z = dz * inv;
  const float s3 = 1.7320508075688772f;    // sqrt(3)
  const float s5 = 2.2360679774997896f;    // sqrt(5)
  const float s15 = 3.8729833462074170f;   // sqrt(15)
  float* Ye = Y + (size_t)e * 9;
  Ye[0] = 1.f;
  Ye[1] = s3 * y; Ye[2] = s3 * z; Ye[3] = s3 * x;
  Ye[4] = s15 * x * y;
  Ye[5] = s15 * y * z;
  Ye[6] = 0.5f * s5 * (3.f * z * z - 1.f);
  Ye[7] = s15 * x * z;
  Ye[8] = 0.5f * s15 * (x * x - y * y);
  // radial bessel * polynomial cutoff (p=5): 1 - 21u^5 + 35u^6 - 15u^7
  float u = rn * 0.1f;
  float u5 = u * u; u5 = u5 * u5 * u;
  float fc = 1.f - 21.f * u5 + 35.f * u5 * u - 15.f * u5 * u * u;
  if (u >= 1.f) fc = 0.f;
  const float pref = 0.44721359549995793f;  // sqrt(2/10)
  float base = 3.14159265358979323846f * rn * 0.1f;
  float* ebe = eb + (size_t)e * 8;
  for (int n = 1; n <= 8; n++)
    ebe[n - 1] = pref * sinf((float)n * base) * inv * fc;
}

// ===================== fused radial MLP (f32 WMMA + TDM) ===================
// 8->64->64->64->OUTW with SiLU between layers. 128 threads = 4 waves,
// block handles 16 edges. W1/W2 are staged into LDS by the Tensor Data Mover,
// overlapped with the layer-0 WMMAs; activations ping-pong through LDS so the
// A/B fragment loads follow the ISA's 16x16x4 f32 WMMA layouts exactly.
__global__ void radial_mlp_kernel(const float* __restrict__ eb,
                                  const float* __restrict__ W0,
                                  const float* __restrict__ W1,
                                  const float* __restrict__ W2,
                                  const float* __restrict__ W3,
                                  float* __restrict__ out, int E, int OUTW) {
  __shared__ float Xa[16][68];
  __shared__ float Xb[16][68];
  __shared__ float Ws1[64][68];
  __shared__ float Ws2[64][68];
  const int tid = threadIdx.x;
  const int wave = tid >> 5, lane = tid & 31;
  const int M = lane & 15, g = lane >> 4, Nc = lane & 15;
  const int e0 = blockIdx.x * 16;

  // kick off TDM weight staging (wave 0; TDM ignores EXEC -> issued once)
  if (tid == 0) {
    tdm_load_w64(W1, &Ws1[0][0]);
    tdm_load_w64(W2, &Ws2[0][0]);
  }

  // ---- layer 0: 8 -> 64 (A/B straight from global, overlaps the TDM) ----
  {
    int c0 = wave * 16;
    v8f acc; for (int i = 0; i < 8; i++) acc[i] = 0.f;
    for (int k0 = 0; k0 < 8; k0 += 4) {
      int ka = k0 + 2 * g;
      int row = e0 + M;
      v2f a, b;
      a[0] = (row < E) ? eb[(size_t)row * 8 + ka] : 0.f;
      a[1] = (row < E) ? eb[(size_t)row * 8 + ka + 1] : 0.f;
      b[0] = W0[ka * 64 + c0 + Nc];
      b[1] = W0[(ka + 1) * 64 + c0 + Nc];
      acc = __builtin_amdgcn_wmma_f32_16x16x4_f32(false, a, false, b, (short)0,
                                                  acc, false, false);
    }
    for (int rr = 0; rr < 8; rr++) Xa[g * 8 + rr][c0 + Nc] = siluf(acc[rr]);
  }
  __builtin_amdgcn_s_wait_tensorcnt(0);  // no-op for waves that issued nothing
  __syncthreads();
  // ---- layer 1: 64 -> 64, Xa -> Xb (B from TDM-staged LDS) ----
  {
    int c0 = wave * 16;
    v8f acc; for (int i = 0; i < 8; i++) acc[i] = 0.f;
    for (int k0 = 0; k0 < 64; k0 += 4) {
      int ka = k0 + 2 * g;
      v2f a, b;
      a[0] = Xa[M][ka]; a[1] = Xa[M][ka + 1];
      b[0] = Ws1[ka][c0 + Nc]; b[1] = Ws1[ka + 1][c0 + Nc];
      acc = __builtin_amdgcn_wmma_f32_16x16x4_f32(false, a, false, b, (short)0,
                                                  acc, false, false);
    }
    for (int rr = 0; rr < 8; rr++) Xb[g * 8 + rr][c0 + Nc] = siluf(acc[rr]);
  }
  __syncthreads();
  // ---- layer 2: 64 -> 64, Xb -> Xa ----
  {
    int c0 = wave * 16;
    v8f acc; for (int i = 0; i < 8; i++) acc[i] = 0.f;
    for (int k0 = 0; k0 < 64; k0 += 4) {
      int ka = k0 + 2 * g;
      v2f a, b;
      a[0] = Xb[M][ka]; a[1] = Xb[M][ka + 1];
      b[0] = Ws2[ka][c0 + Nc]; b[1] = Ws2[ka + 1][c0 + Nc];
      acc = __builtin_amdgcn_wmma_f32_16x16x4_f32(false, a, false, b, (short)0,
                                                  acc, false, false);
    }
    for (int rr = 0; rr < 8; rr++) Xa[g * 8 + rr][c0 + Nc] = siluf(acc[rr]);
  }
  __syncthreads();
  // ---- layer 3: 64 -> OUTW (no activation; W3 too wide for LDS at 704) ----
  for (int c0 = wave * 16; c0 < OUTW; c0 += 64) {
    v8f acc; for (int i = 0; i < 8; i++) acc[i] = 0.f;
    for (int k0 = 0; k0 < 64; k0 += 4) {
      int ka = k0 + 2 * g;
      v2f a, b;
      a[0] = Xa[M][ka]; a[1] = Xa[M][ka + 1];
      b[0] = W3[ka * OUTW + c0 + Nc]; b[1] = W3[(ka + 1) * OUTW + c0 + Nc];
      acc = __builtin_amdgcn_wmma_f32_16x16x4_f32(false, a, false, b, (short)0,
                                                  acc, false, false);
    }
    for (int rr = 0; rr < 8; rr++) {
      int row = e0 + g * 8 + rr;
      if (row < E) out[(size_t)row * OUTW + c0 + Nc] = acc[rr];
    }
  }
}

// ============== generic per_l_linear GEMM (f32 WMMA + TDM/async) ===========
// out[n][f][d] (+= addin) = sum_c x[n][c][d] * W[c][f], rows R = Nn*D.
// x element at ((n*64+c)*ldx + offx + d); out/addin use (ldo, offo).
// W is staged to LDS by the TDM; the strided A gather goes through
// global_load_async_to_lds_b32 (ASYNCcnt) instead of VGPR staging.
__global__ void perl_linear_kernel(const float* __restrict__ x,
                                   const float* __restrict__ W,
                                   const float* __restrict__ addin,
                                   float* __restrict__ out, int Nn, int D,
                                   int ldx, int offx, int ldo, int offo,
                                   int act) {
  __shared__ float Xs[16][68];
  __shared__ float Ws[64][68];
  const int tid = threadIdx.x;
  const int R = Nn * D;
  const int r0 = blockIdx.x * 16;
  // TDM: stage 64x64 W into LDS (padded row stride 68), issued once.
  if (tid == 0) tdm_load_w64(W, &Ws[0][0]);
  // async gather of the (n,d)-strided A tile straight into LDS
  const unsigned xs_base = (unsigned)(uintptr_t)&Xs[0][0];
  for (int idx = tid; idx < 1024; idx += blockDim.x) {
    int rr = idx >> 6, c = idx & 63;
    int r = r0 + rr;
    int rc = (r < R) ? r : (R - 1);  // clamp; tail zeroed below
    int n = rc / D, d = rc - n * D;
    async_gather_b32(xs_base + (unsigned)(rr * 68 + c) * 4u,
                     x + ((size_t)n * 64 + c) * ldx + offx + d);
  }
  async_wait0();
  __builtin_amdgcn_s_wait_tensorcnt(0);
  __syncthreads();
  if (r0 + 16 > R) {  // block-uniform tail fixup
    for (int idx = tid; idx < 1024; idx += blockDim.x) {
      int rr = idx >> 6, c = idx & 63;
      if (r0 + rr >= R) Xs[rr][c] = 0.f;
    }
    __syncthreads();
  }
  const int wave = tid >> 5, lane = tid & 31;
  const int M = lane & 15, g = lane >> 4, Nc = lane & 15;
  const int c0 = wave * 16;
  v8f acc; for (int i = 0; i < 8; i++) acc[i] = 0.f;
  for (int k0 = 0; k0 < 64; k0 += 4) {
    int ka = k0 + 2 * g;
    v2f a, b;
    a[0] = Xs[M][ka]; a[1] = Xs[M][ka + 1];
    b[0] = Ws[ka][c0 + Nc]; b[1] = Ws[ka + 1][c0 + Nc];
    acc = __builtin_amdgcn_wmma_f32_16x16x4_f32(false, a, false, b, (short)0,
                                                acc, false, false);
  }
  for (int rr = 0; rr < 8; rr++) {
    int r = r0 + g * 8 + rr;
    if (r < R) {
      int n = r / D, d = r - n * D;
      size_t oidx = ((size_t)n * 64 + c0 + Nc) * ldo + offo + d;
      float v = acc[rr];
      if (addin) v += addin[oidx];
      if (act) v = siluf(v);
      out[oidx] = v;
    }
  }
}

// ==================== conv_tp #1 (paths (0,l,l)) ==========================
__global__ void msg1_kernel(const float* __restrict__ Y,
                            const float* __restrict__ rw1,
                            const int* __restrict__ rcv,
                            const float* __restrict__ hvec,
                            const float* __restrict__ CG,
                            float* __restrict__ m1, int E) {
  int gid = blockIdx.x * blockDim.x + threadIdx.x;
  int e = gid >> 6, c = gid & 63;
  if (e >= E) return;
  float hv = hvec[c];
  const float* Ye = Y + (size_t)e * 9;
  int rv = rcv[e];
  int off = 0;
  for (int l = 0; l < 3; l++) {
    int dl = 2 * l + 1;
    float w = rw1[(size_t)e * 192 + l * 64 + c] * hv;
    const float* cg = CG + (((0 * 3 + l) * 3 + l) * 125);
    for (int k = 0; k < dl; k++) {
      float s = 0.f;
      for (int m = 0; m < dl; m++) s += cg[m * 5 + k] * Ye[off + m];
      atomicAdd(&m1[((size_t)rv * 64 + c) * 9 + off + k], w * s);
    }
    off += dl;
  }
}

// ==================== conv_tp #2 (all 11 CG paths) ========================
__global__ void msg2_kernel(const float* __restrict__ Y,
                            const float* __restrict__ rw2,
                            const float* __restrict__ h2,
                            const int* __restrict__ snd,
                            const int* __restrict__ rcv,
                            const float* __restrict__ CG,
                            float* __restrict__ m2, int E) {
  int gid = blockIdx.x * blockDim.x + threadIdx.x;
  int e = gid >> 6, c = gid & 63;
  if (e >= E) return;
  const int loff[3] = {0, 1, 4};
  float xs[9], Ye[9], acc[9];
  const float* hb = h2 + ((size_t)snd[e] * 64 + c) * 9;
  const float* yb = Y + (size_t)e * 9;
  for (int i = 0; i < 9; i++) { xs[i] = hb[i]; Ye[i] = yb[i]; acc[i] = 0.f; }
  int p = 0;
  for (int l1 = 0; l1 < 3; l1++)
    for (int l2 = 0; l2 < 3; l2++)
      for (int l3 = 0; l3 < 3; l3++) {
        if (!validl(l1, l2, l3)) continue;
        float w = rw2[(size_t)e * 704 + p * 64 + c];
        p++;
        const float* cg = CG + (((l1 * 3 + l2) * 3 + l3) * 125);
        for (int k = 0; k < 2 * l3 + 1; k++) {
          float s = 0.f;
          for (int i = 0; i < 2 * l1 + 1; i++)
            for (int j = 0; j < 2 * l2 + 1; j++)
              s += cg[(i * 5 + j) * 5 + k] * xs[loff[l1] + i] * Ye[loff[l2] + j];
          acc[loff[l3] + k] += w * s;
        }
      }
  int rv = rcv[e];
  for (int k = 0; k < 9; k++)
    atomicAdd(&m2[((size_t)rv * 64 + c) * 9 + k], acc[k]);
}

// ===================== symmetric contraction (1st/2nd) =====================
__global__ void sym_contract_kernel(const float* __restrict__ xin,
                                    const float* w1_0, const float* w1_1,
                                    const float* w1_2, const float* w2_0,
                                    const float* w2_1, const float* w2_2,
                                    const float* w3_0, const float* w3_1,
                                    const float* w3_2,
                                    const float* __restrict__ CG,
                                    float* __restrict__ outp, int Nn,
                                    int only0) {
  int gid = blockIdx.x * blockDim.x + threadIdx.x;
  int n = gid >> 6, c = gid & 63;
  if (n >= Nn) return;
  const int loff[3] = {0, 1, 4};
  float x[9];
  const float* xb = xin + ((size_t)n * 64 + c) * 9;
  for (int i = 0; i < 9; i++) x[i] = xb[i];
  int nlo = only0 ? 1 : 3;
  for (int lo = 0; lo < nlo; lo++) {
    const float* w1 = (lo == 0) ? w1_0 : ((lo == 1) ? w1_1 : w1_2);
    const float* w2 = (lo == 0) ? w2_0 : ((lo == 1) ? w2_1 : w2_2);
    const float* w3 = (lo == 0) ? w3_0 : ((lo == 1) ? w3_1 : w3_2);
    int dlo = 2 * lo + 1;
    float acc[5];
    float w1c = w1[c];
    for (int k = 0; k < dlo; k++) acc[k] = w1c * x[loff[lo] + k];
    // order-2 paths (l1-major, mirrors the reference list construction)
    int i2 = 0;
    for (int l1 = 0; l1 < 3; l1++)
      for (int l2 = 0; l2 < 3; l2++) {
        if (!validl(l1, l2, lo)) continue;
        float wc = w2[i2 * 64 + c];
        i2++;
        const float* cg = CG + (((l1 * 3 + l2) * 3 + lo) * 125);
        for (int k = 0; k < dlo; k++) {
          float s = 0.f;
          for (int i = 0; i < 2 * l1 + 1; i++)
            for (int j = 0; j < 2 * l2 + 1; j++)
              s += cg[(i * 5 + j) * 5 + k] * x[loff[l1] + i] * x[loff[l2] + j];
          acc[k] += wc * s;
        }
      }
    // order-3 paths
    int i3 = 0;
    for (int l1 = 0; l1 < 3; l1++)
      for (int l2 = 0; l2 < 3; l2++)
        for (int L = 0; L < 3; L++)
          for (int l3 = 0; l3 < 3; l3++) {
            if (!(validl(l1, l2, L) && validl(L, l3, lo))) continue;
            float wc = w3[i3 * 64 + c];
            i3++;
            float t[5];
            int dL = 2 * L + 1;
            const float* cgA = CG + (((l1 * 3 + l2) * 3 + L) * 125);
            for (int k = 0; k < dL; k++) {
              float s = 0.f;
              for (int i = 0; i < 2 * l1 + 1; i++)
                for (int j = 0; j < 2 * l2 + 1; j++)
                  s += cgA[(i * 5 + j) * 5 + k] * x[loff[l1] + i] * x[loff[l2] + j];
              t[k] = s;
            }
            const float* cgB = CG + (((L * 3 + l3) * 3 + lo) * 125);
            for (int m = 0; m < dlo; m++) {
              float s = 0.f;
              for (int k = 0; k < dL; k++)
                for (int j = 0; j < 2 * l3 + 1; j++)
                  s += cgB[(k * 5 + j) * 5 + m] * t[k] * x[loff[l3] + j];
              acc[m] += wc * s;
            }
          }
    if (only0) {
      outp[(size_t)n * 64 + c] = acc[0];
    } else {
      float* ob = outp + ((size_t)n * 64 + c) * 9 + loff[lo];
      for (int k = 0; k < dlo; k++) ob[k] = acc[k];
    }
  }
}

// =========================== readout helpers ==============================
__global__ void node_dot_scatter_kernel(const float* __restrict__ feat, int ld,
                                        const float* __restrict__ wvec,
                                        const int* __restrict__ batch,
                                        float* __restrict__ acc, int Nn) {
  int n = blockIdx.x * blockDim.x + threadIdx.x;
  if (n >= Nn) return;
  const float* fb = feat + (size_t)n * 64 * ld;
  float s = 0.f;
  for (int f = 0; f < 64; f++) s += fb[(size_t)f * ld] * wvec[f];
  atomicAdd(&acc[batch[n]], s);
}

__global__ void count_kernel(const int* __restrict__ batch,
                             float* __restrict__ cnt, int Nn) {
  int n = blockIdx.x * blockDim.x + threadIdx.x;
  if (n >= Nn) return;
  atomicAdd(&cnt[batch[n]], 1.0f);
}

__global__ void finalize_kernel(const float* cnt, const float* dot_ae,
                                const float* e1, const float* e2,
                                float* __restrict__ out) {
  int g = threadIdx.x;
  if (g < NGRAPH) out[g] = cnt[g] * dot_ae[0] + e1[g] + e2[g];
}

// =============================== launcher =================================
extern "C" void kernel_launch(void* const* d_in, const int* in_sizes, int n_in,
                              void* d_out, int out_size, void* d_ws,
                              size_t ws_size, hipStream_t stream) {
  (void)in_sizes; (void)n_in; (void)out_size; (void)ws_size;
  const int NN = NNODES, NE = NEDGES;
  const float* pos = (const float*)d_in[0];
  auto p = [&](int i) { return (const float*)d_in[1 + i]; };
  const int* ei = (const int*)d_in[45];
  const int* batch = (const int*)d_in[46];

  float* ws = (float*)d_ws;
  size_t o = 0;
  float* CGt  = ws + o; o += 27 * 125;
  float* hvec = ws + o; o += 64;
  float* dota = ws + o; o += 16;
  float* cnt  = ws + o; o += 16;
  float* e1a  = ws + o; o += 16;
  float* e2a  = ws + o; o += 16;
  float* weff = ws + o; o += 3 * 4096;
  float* Yb   = ws + o; o += (size_t)NE * 9;
  float* ebb  = ws + o; o += (size_t)NE * 8;
  float* rw1  = ws + o; o += (size_t)NE * 192;
  float* rw2  = ws + o; o += (size_t)NE * 704;
  float* m1   = ws + o; o += (size_t)NN * 64 * 9;
  float* mlin = ws + o; o += (size_t)NN * 64 * 9;
  float* bb   = ws + o; o += (size_t)NN * 64 * 9;
  float* hf   = ws + o; o += (size_t)NN * 64 * 9;
  float* h2   = ws + o; o += (size_t)NN * 64 * 9;
  float* m2   = ws + o; o += (size_t)NN * 64 * 9;
  float* m2l  = ws + o; o += (size_t)NN * 64 * 9;
  float* b2   = ws + o; o += (size_t)NN * 64;
  float* scb  = ws + o; o += (size_t)NN * 64;
  float* sb   = ws + o; o += (size_t)NN * 64;
  float* t2   = ws + o; o += (size_t)NN * 64;

  const int Dl[3] = {1, 3, 5};
  const int OF[3] = {0, 1, 4};

  // zero accumulators (graph-capturable memset nodes)
  hipMemsetAsync(cnt, 0, 48 * sizeof(float), stream);  // cnt,e1a,e2a contiguous
  hipMemsetAsync(m1, 0, (size_t)NN * 64 * 9 * sizeof(float), stream);
  hipMemsetAsync(m2, 0, (size_t)NN * 64 * 9 * sizeof(float), stream);

  cg_init_kernel<<<1, 32, 0, stream>>>(CGt);
  node_embed_kernel<<<1, 64, 0, stream>>>(p(0), p(2), p(1), hvec, dota);
  edge_geom_kernel<<<(NE + 255) / 256, 256, 0, stream>>>(pos, ei, Yb, ebb, NE);

  // rw1 = mlp(eb, rmlp1) : out width 3*64
  radial_mlp_kernel<<<(NE + 15) / 16, 128, 0, stream>>>(ebb, p(3), p(4), p(5),
                                                        p(6), rw1, NE, 192);
  // conv_tp #1 + segment_sum -> m1
  msg1_kernel<<<(NE * 64) / 256, 256, 0, stream>>>(Yb, rw1, ei + NE, hvec, CGt,
                                                   m1, NE);
  // Weff_l = lin1_l @ skip1_l (composition of the two per_l_linear layers)
  for (int l = 0; l < 3; l++)
    perl_linear_kernel<<<4, 128, 0, stream>>>(p(7 + l), p(10 + l), nullptr,
                                              weff + l * 4096, 64, 1, 1, 0, 1,
                                              0, 0);
  for (int l = 0; l < 3; l++)
    perl_linear_kernel<<<(NN * Dl[l] + 15) / 16, 128, 0, stream>>>(
        m1, weff + l * 4096, nullptr, mlin, NN, Dl[l], 9, OF[l], 9, OF[l], 0);
  // b = sym_contract(m, p1_*)
  sym_contract_kernel<<<(NN * 64) / 256, 256, 0, stream>>>(
      mlin, p(13), p(14), p(15), p(16), p(17), p(18), p(19), p(20), p(21), CGt,
      bb, NN, 0);
  // hf = per_l_linear(b, p1_lin)
  for (int l = 0; l < 3; l++)
    perl_linear_kernel<<<(NN * Dl[l] + 15) / 16, 128, 0, stream>>>(
        bb, p(22 + l), nullptr, hf, NN, Dl[l], 9, OF[l], 9, OF[l], 0);
  // e1 = segment_sum(hf0 @ read1)
  node_dot_scatter_kernel<<<(NN + 255) / 256, 256, 0, stream>>>(hf, 9, p(25),
                                                                batch, e1a, NN);
  // sc = hf0 @ sc2
  perl_linear_kernel<<<(NN + 15) / 16, 128, 0, stream>>>(hf, p(26), nullptr,
                                                         scb, NN, 1, 9, 0, 1,
                                                         0, 0);
  // h2 = per_l_linear(hf, lin_up2)
  for (int l = 0; l < 3; l++)
    perl_linear_kernel<<<(NN * Dl[l] + 15) / 16, 128, 0, stream>>>(
        hf, p(27 + l), nullptr, h2, NN, Dl[l], 9, OF[l], 9, OF[l], 0);
  // rw2 = mlp(eb, rmlp2) : out width 11*64
  radial_mlp_kernel<<<(NE + 15) / 16, 128, 0, stream>>>(ebb, p(30), p(31),
                                                        p(32), p(33), rw2, NE,
                                                        704);
  // conv_tp #2 + segment_sum -> m2
  msg2_kernel<<<(NE * 64) / 256, 256, 0, stream>>>(Yb, rw2, h2, ei, ei + NE,
                                                   CGt, m2, NE);
  // m2 = per_l_linear(m2, lin2)
  for (int l = 0; l < 3; l++)
    perl_linear_kernel<<<(NN * Dl[l] + 15) / 16, 128, 0, stream>>>(
        m2, p(34 + l), nullptr, m2l, NN, Dl[l], 9, OF[l], 9, OF[l], 0);
  // b2 = sym_contract(m2, p2_*, louts=[0])
  sym_contract_kernel<<<(NN * 64) / 256, 256, 0, stream>>>(
      m2l, p(37), p(37), p(37), p(38), p(38), p(38), p(39), p(39), p(39), CGt,
      b2, NN, 1);
  // s = b2 @ p2_lin + sc
  perl_linear_kernel<<<(NN + 15) / 16, 128, 0, stream>>>(b2, p(40), scb, sb,
                                                         NN, 1, 1, 0, 1, 0, 0);
  // t2 = silu(s @ read2_w1)
  perl_linear_kernel<<<(NN + 15) / 16, 128, 0, stream>>>(sb, p(41), nullptr,
                                                         t2, NN, 1, 1, 0, 1, 0,
                                                         1);
  // e2 = segment_sum(t2 @ read2_w2)
  node_dot_scatter_kernel<<<(NN + 255) / 256, 256, 0, stream>>>(t2, 1, p(42),
                                                                batch, e2a, NN);
  // e0 = count[g] * dot(node_embed, ae_w)
  count_kernel<<<(NN + 255) / 256, 256, 0, stream>>>(batch, cnt, NN);
  finalize_kernel<<<1, 16, 0, stream>>>(cnt, dota, e1a, e2a, (float*)d_out);
}